// BinarizeSauvolaLayer_84155589198484
// MI455X (gfx1250) — compile-verified
//
#include <hip/hip_runtime.h>
#include <stdint.h>

// Sauvola binarization, MI455X (gfx1250).
// Bandwidth-bound (~320 MB => ~14us @ 23.3 TB/s). The 51-tap window sums stay
// on VALU running sums (a banded-matrix WMMA formulation would inflate FLOPs
// ~70x for zero byte savings). CDNA5-specific paths used:
//   - TDM tensor_load_to_lds + s_wait_tensorcnt  (row DMA into LDS)
//   - global_load_async_to_lds_b32/b128 + s_wait_asynccnt (LDS ring pipeline)
//   - V_WMMA_F32_16X16X4_F32 for the RGB->gray dot (matrix pipe is idle;
//     D = A(16px x 4ch) x B(4ch x 16, weights broadcast), ~1us total, hidden)
//
// Workspace layout (requires ~96MB + 256B):
//   [0..8)            : uint32 minBits, maxBits
//   +256              : gray   (8M f32, 32MB)
//   +256+32MB         : sumW_g (8M f32, 32MB)   51-tap horizontal sum of g
//   +256+64MB         : sumW_g2(8M f32, 32MB)   51-tap horizontal sum of g^2

#define BATCH 8
#define IMG_H 1024
#define IMG_W 1024
#define NPIX  (BATCH * IMG_H * IMG_W)
#define WIN   51
#define HALFW 25
#define KSAU  0.2f

#define SEG    64   // rows per column-pass block
#define PDEPTH 3    // async prefetch depth (row-pairs in flight)
#define RING   56   // LDS ring rows (>= 51 + PDEPTH + 2 slack)

// ---------------- CDNA5 async / TDM / WMMA helpers (probe-guarded) ---------
typedef unsigned int u32x4 __attribute__((ext_vector_type(4)));
typedef int          i32x4 __attribute__((ext_vector_type(4)));
typedef int          i32x8 __attribute__((ext_vector_type(8)));
typedef float        v2f   __attribute__((ext_vector_type(2)));
typedef float        v8f   __attribute__((ext_vector_type(8)));

// Builtin-expected pointer types (LLVM sig "vi*1i*3IiIi" / "vV4i*1V4i*3IiIi"):
typedef __attribute__((address_space(1))) int    g_i32;
typedef __attribute__((address_space(3))) int    l_i32;
typedef __attribute__((address_space(1))) i32x4  g_i32x4;
typedef __attribute__((address_space(3))) i32x4  l_i32x4;

#if __has_builtin(__builtin_amdgcn_global_load_async_to_lds_b32)
__device__ __forceinline__ void async_copy_b32(const void* g, void* l) {
  __builtin_amdgcn_global_load_async_to_lds_b32(
      (g_i32*)(uintptr_t)g,
      (l_i32*)(uint32_t)(uintptr_t)l,   // low 32 bits = LDS byte offset
      0, 0);
}
__device__ __forceinline__ void async_copy_b128(const void* g, void* l) {
  __builtin_amdgcn_global_load_async_to_lds_b128(
      (g_i32x4*)(uintptr_t)g,
      (l_i32x4*)(uint32_t)(uintptr_t)l,
      0, 0);
}
#else
__device__ __forceinline__ void async_copy_b32(const void* g, void* l) {
  asm volatile("global_load_async_to_lds_b32 %0, %1, off"
               :: "v"((uint32_t)(uintptr_t)l), "v"(g) : "memory");
}
__device__ __forceinline__ void async_copy_b128(const void* g, void* l) {
  asm volatile("global_load_async_to_lds_b128 %0, %1, off"
               :: "v"((uint32_t)(uintptr_t)l), "v"(g) : "memory");
}
#endif

#if __has_builtin(__builtin_amdgcn_s_wait_asynccnt)
#define WAIT_ASYNCCNT(n) __builtin_amdgcn_s_wait_asynccnt(n)
#else
#define WAIT_ASYNCCNT(n) asm volatile("s_wait_asynccnt %0" :: "i"(n) : "memory")
#endif

#if __has_builtin(__builtin_amdgcn_wmma_f32_16x16x4_f32)
#define HAVE_WMMA_F32 1
#else
#define HAVE_WMMA_F32 0
#endif

#if __has_builtin(__builtin_amdgcn_tensor_load_to_lds)
#define HAVE_TDM 1
// DMA one contiguous row of nElems f32 from global into LDS via the TDM.
// D# packing per CDNA5 ISA (2D tensor, tile = 1 x nElems).
__device__ __forceinline__ void tdm_load_row(const void* gsrc, void* ldst, int nElems) {
  uint64_t ga = (uint64_t)(uintptr_t)gsrc;
  uint32_t la = (uint32_t)(uintptr_t)ldst;
  u32x4 g0;
  g0[0] = 1u;                                    // count=1, user descriptor
  g0[1] = la;                                    // lds_addr (bytes)
  g0[2] = (uint32_t)ga;                          // global_addr[31:0]
  g0[3] = (uint32_t)(ga >> 32) | (2u << 30);     // global_addr[56:32] | type=2
  uint32_t d0 = (uint32_t)nElems;
  i32x8 g1;
  g1[0] = (int)(2u << 16);                       // data_size = 4B; mask=0
  g1[1] = (int)((d0 & 0xFFFFu) << 16);           // tensor_dim0[15:0]
  g1[2] = (int)((d0 >> 16) | (1u << 16));        // tensor_dim0[31:16] | tensor_dim1=1
  g1[3] = (int)((d0 & 0xFFFFu) << 16);           // tile_dim0 = nElems
  g1[4] = 1;                                     // tile_dim1=1, tile_dim2=0
  g1[5] = (int)d0;                               // tensor_dim0_stride lo
  g1[6] = 0;                                     // stride hi | dim1_stride lo
  g1[7] = 0;
  i32x4 g2; g2[0] = 1; g2[1] = 0; g2[2] = 0; g2[3] = 0;  // tensor_dim2=1
  i32x4 g3; g3[0] = 0; g3[1] = 0; g3[2] = 0; g3[3] = 0;
#if defined(__clang_major__) && (__clang_major__ >= 23)
  i32x8 g4; g4[0]=0; g4[1]=0; g4[2]=0; g4[3]=0; g4[4]=0; g4[5]=0; g4[6]=0; g4[7]=0;
  __builtin_amdgcn_tensor_load_to_lds(g0, g1, g2, g3, g4, 0);
#else
  __builtin_amdgcn_tensor_load_to_lds(g0, g1, g2, g3, 0);
#endif
}
#else
#define HAVE_TDM 0
#endif

__device__ __forceinline__ int reflect_i(int i, int n) {
  i = (i < 0) ? -i : i;                 // jnp.pad 'reflect': -1 -> 1
  i = (i >= n) ? (2 * n - 2 - i) : i;   // n -> n-2
  return i;
}

// ---------------------------------------------------------------------------
__global__ void init_mm(unsigned* mm) {
  mm[0] = 0x7F800000u;   // +inf  (gray >= 0, bit order == float order)
  mm[1] = 0u;            // 0.0f
}

// Pass A: one block per (b,h) image row.
// RGB row -> LDS (TDM), gray via WMMA (LDS + global), min/max atomics,
// 51-tap horizontal running sums of g and g*g -> global.
__global__ __launch_bounds__(256) void sauvola_rows(
    const float* __restrict__ in, float* __restrict__ gray,
    float* __restrict__ sg, float* __restrict__ sg2, unsigned* __restrict__ mm) {
  __shared__ float ldsRGB[IMG_W * 3];
  __shared__ float ldsG[IMG_W];
  __shared__ float redmin[8], redmax[8];

  const int row = blockIdx.x;                       // 0 .. B*H-1
  const float* src = in + (size_t)row * (IMG_W * 3);
  const int t = threadIdx.x;

#if HAVE_TDM
  if (t < 32) {                                     // wave 0 only (uniform branch)
    tdm_load_row(src, ldsRGB, IMG_W * 3);
#if __has_builtin(__builtin_amdgcn_s_wait_tensorcnt)
    __builtin_amdgcn_s_wait_tensorcnt(0);
#else
    asm volatile("s_wait_tensorcnt 0x0" ::: "memory");
#endif
  }
#else
  // Fallback: 3x 16B async copies per lane (48B/lane, 12KB/row).
  async_copy_b128(src + t * 12 + 0, &ldsRGB[t * 12 + 0]);
  async_copy_b128(src + t * 12 + 4, &ldsRGB[t * 12 + 4]);
  async_copy_b128(src + t * 12 + 8, &ldsRGB[t * 12 + 8]);
  WAIT_ASYNCCNT(0);
#endif
  __syncthreads();

  const int p0 = t * 4;
  float lmin, lmax;

#if HAVE_WMMA_F32
  // RGB->gray on the (otherwise idle) matrix pipe.
  // D(16x16) = A(16x4) x B(4x16): A rows = 16 pixels, K = {r,g,b,pad};
  // B[k][n] = weight_k for all n => D[m][n] = gray(pixel m), all columns equal.
  // A f32 16x4 layout: lanes 0-15 hold K=0 (v0) / K=1 (v1) for row M=lane;
  //                    lanes 16-31 hold K=2 (v0) / K=3 (v1) for row M=lane-16.
  // B 4x16 mirrors this with N=lane%16.
  {
    const int lane = t & 31;
    const int wv = t >> 5;                 // 8 waves, 128 px each
    const int Lm = lane & 15;
    const bool lo = (lane < 16);
    v2f bmat;
    bmat.x = lo ? 0.2989f : 0.1140f;       // K=0: w_r   | K=2: w_b
    bmat.y = lo ? 0.5870f : 0.0f;          // K=1: w_g   | K=3: pad
#pragma unroll
    for (int it = 0; it < 8; ++it) {       // 16 pixels per WMMA
      const int q0 = wv * 128 + it * 16;
      const int b3 = (q0 + Lm) * 3;
      v2f a;
      a.x = lo ? ldsRGB[b3 + 0] : ldsRGB[b3 + 2];
      a.y = lo ? ldsRGB[b3 + 1] : 0.0f;
      v8f c = {};
      v8f d = __builtin_amdgcn_wmma_f32_16x16x4_f32(
          false, a, false, bmat, (short)0, c, false, false);
      // D VGPR i: lanes 0-15 = pixel q0+i, lanes 16-31 = pixel q0+8+i.
      if (lane == 0) {
#pragma unroll
        for (int i = 0; i < 8; ++i) ldsG[q0 + i] = d[i];
      } else if (lane == 16) {
#pragma unroll
        for (int i = 0; i < 8; ++i) ldsG[q0 + 8 + i] = d[i];
      }
    }
  }
  __syncthreads();
  {
    float4 gv = *(const float4*)&ldsG[p0];
    *(float4*)(gray + (size_t)row * IMG_W + p0) = gv;
    lmin = fminf(fminf(gv.x, gv.y), fminf(gv.z, gv.w));
    lmax = fmaxf(fmaxf(gv.x, gv.y), fmaxf(gv.z, gv.w));
  }
#else
  // Scalar fallback path.
  lmin = 3.4e38f; lmax = -3.4e38f;
  float g4[4];
#pragma unroll
  for (int i = 0; i < 4; ++i) {
    const float* px = &ldsRGB[(p0 + i) * 3];
    float g = 0.2989f * px[0] + 0.5870f * px[1] + 0.1140f * px[2];
    ldsG[p0 + i] = g;
    g4[i] = g;
    lmin = fminf(lmin, g);
    lmax = fmaxf(lmax, g);
  }
  *(float4*)(gray + (size_t)row * IMG_W + p0) = make_float4(g4[0], g4[1], g4[2], g4[3]);
#endif

  // wave32 tree reduce, then cross-wave via LDS, then global atomics
#pragma unroll
  for (int off = 16; off > 0; off >>= 1) {
    lmin = fminf(lmin, __shfl_xor(lmin, off));
    lmax = fmaxf(lmax, __shfl_xor(lmax, off));
  }
  if ((t & 31) == 0) { redmin[t >> 5] = lmin; redmax[t >> 5] = lmax; }
  __syncthreads();
  if (t == 0) {
    float bmin = redmin[0], bmax = redmax[0];
#pragma unroll
    for (int i = 1; i < 8; ++i) { bmin = fminf(bmin, redmin[i]); bmax = fmaxf(bmax, redmax[i]); }
    atomicMin(&mm[0], __float_as_uint(bmin));
    atomicMax(&mm[1], __float_as_uint(bmax));
  }

  // horizontal 51-tap running sums (reflect padding), 4 px/thread
  float s1 = 0.f, s2 = 0.f;
  for (int k = -HALFW; k <= HALFW; ++k) {
    float v = ldsG[reflect_i(p0 + k, IMG_W)];
    s1 += v;
    s2 += v * v;
  }
  float o1[4], o2[4];
  o1[0] = s1; o2[0] = s2;
#pragma unroll
  for (int i = 1; i < 4; ++i) {
    float va = ldsG[reflect_i(p0 + i + HALFW, IMG_W)];
    float vs = ldsG[reflect_i(p0 + i - HALFW - 1, IMG_W)];
    s1 += va - vs;
    s2 += va * va - vs * vs;
    o1[i] = s1; o2[i] = s2;
  }
  size_t o = (size_t)row * IMG_W + p0;
  *(float4*)(sg + o)  = make_float4(o1[0], o1[1], o1[2], o1[3]);
  *(float4*)(sg2 + o) = make_float4(o2[0], o2[1], o2[2], o2[3]);
}

// Pass B: vertical 51-tap running sums via an async-filled LDS ring
// (RING rows x 256 cols for each of sumW_g / sumW_g2 = 112KB), prefetch
// depth PDEPTH row-pairs, then threshold + compare + store.
__global__ __launch_bounds__(256) void sauvola_cols(
    const float* __restrict__ gray, const float* __restrict__ sg,
    const float* __restrict__ sg2, const unsigned* __restrict__ mm,
    float* __restrict__ out) {
  extern __shared__ float lds[];
  float* rg  = lds;                  // [RING][256]
  float* rg2 = lds + RING * 256;     // [RING][256]

  const int t = threadIdx.x;
  const int stripes = IMG_W / 256;   // 4
  const int segs    = IMG_H / SEG;   // 16
  const int bw = blockIdx.x % stripes;
  const int hs = (blockIdx.x / stripes) % segs;
  const int b  = blockIdx.x / (stripes * segs);
  const int h0 = hs * SEG;
  const int c  = bw * 256 + t;
  const size_t img = (size_t)b * (IMG_H * IMG_W);

  const float gmin = __uint_as_float(mm[0]);
  const float gmax = __uint_as_float(mm[1]);
  const float invr = 1.0f / (0.5f * (gmax - gmin));
  const float invWW = 1.0f / (float)(WIN * WIN);

  // fill ring: rows idx 0 .. 50+PDEPTH (logical row j = h0-25+idx, reflected)
  for (int idx = 0; idx <= (WIN - 1) + PDEPTH; ++idx) {
    int j = reflect_i(h0 - HALFW + idx, IMG_H);
    size_t go = img + (size_t)j * IMG_W + c;
    async_copy_b32(sg  + go, &rg [idx * 256 + t]);
    async_copy_b32(sg2 + go, &rg2[idx * 256 + t]);
  }
  WAIT_ASYNCCNT(2 * PDEPTH);         // rows idx 0..50 have landed

  float S1 = 0.f, S2 = 0.f;
  for (int idx = 0; idx < WIN; ++idx) {
    S1 += rg[idx * 256 + t];
    S2 += rg2[idx * 256 + t];
  }
  {
    float m = S1 * invWW, m2 = S2 * invWW;
    float s = sqrtf(fmaxf(m2 - m * m, 0.f));
    float th = m * (1.f + KSAU * (s * invr - 1.f));
    size_t o = img + (size_t)h0 * IMG_W + c;
    out[o] = (gray[o] > th) ? 1.f : 0.f;
  }

  int slot_pf  = WIN + PDEPTH;       // next ring slot to fill (idx 54)
  int slot_add = WIN;                // idx of row h+25 for h = h0+1
  int slot_sub = 0;                  // idx of row h-26 for h = h0+1
  for (int h = h0 + 1; h < h0 + SEG; ++h) {
    int jp = reflect_i(h + HALFW + PDEPTH, IMG_H);
    size_t gp = img + (size_t)jp * IMG_W + c;
    async_copy_b32(sg  + gp, &rg [slot_pf * 256 + t]);
    async_copy_b32(sg2 + gp, &rg2[slot_pf * 256 + t]);
    slot_pf = (slot_pf + 1 == RING) ? 0 : slot_pf + 1;
    WAIT_ASYNCCNT(2 * PDEPTH);       // guarantees the add-row is resident

    S1 += rg [slot_add * 256 + t] - rg [slot_sub * 256 + t];
    S2 += rg2[slot_add * 256 + t] - rg2[slot_sub * 256 + t];
    slot_add = (slot_add + 1 == RING) ? 0 : slot_add + 1;
    slot_sub = (slot_sub + 1 == RING) ? 0 : slot_sub + 1;

    float m = S1 * invWW, m2 = S2 * invWW;
    float s = sqrtf(fmaxf(m2 - m * m, 0.f));
    float th = m * (1.f + KSAU * (s * invr - 1.f));
    size_t o = img + (size_t)h * IMG_W + c;
    out[o] = (gray[o] > th) ? 1.f : 0.f;
  }
}

// ---------------------------------------------------------------------------
extern "C" void kernel_launch(void* const* d_in, const int* in_sizes, int n_in,
                              void* d_out, int out_size, void* d_ws, size_t ws_size,
                              hipStream_t stream) {
  (void)in_sizes; (void)n_in; (void)out_size; (void)ws_size;
  const float* in = (const float*)d_in[0];
  float* out = (float*)d_out;
  unsigned* mm = (unsigned*)d_ws;
  float* gray = (float*)((char*)d_ws + 256);
  float* sg   = gray + (size_t)NPIX;
  float* sg2  = sg + (size_t)NPIX;

  init_mm<<<1, 1, 0, stream>>>(mm);
  sauvola_rows<<<BATCH * IMG_H, 256, 0, stream>>>(in, gray, sg, sg2, mm);
  const size_t ldsBytes = (size_t)RING * 256 * 2 * sizeof(float);  // 112KB of 320KB
  sauvola_cols<<<BATCH * (IMG_W / 256) * (IMG_H / SEG), 256, ldsBytes, stream>>>(
      gray, sg, sg2, mm, out);
}